// EctLayer_32993938768454
// MI455X (gfx1250) — compile-verified
//
#include <hip/hip_runtime.h>
#include <hip/hip_bf16.h>
#include <stdint.h>

// Problem constants (match the reference)
#define N_NODES   50000
#define N_EDGES   100000
#define N_GRAPHS  128
#define N_FEAT    3
#define N_T       32      // directions
#define N_S       32      // thresholds

typedef __attribute__((ext_vector_type(2))) float v2f;
typedef __attribute__((ext_vector_type(8))) float v8f;
typedef __attribute__((ext_vector_type(4))) unsigned int u32x4;
typedef __attribute__((ext_vector_type(8))) int i32x8;
typedef __attribute__((ext_vector_type(4))) int i32x4;

#if defined(__has_builtin)
#if __has_builtin(__builtin_amdgcn_tensor_load_to_lds) && __has_builtin(__builtin_amdgcn_s_wait_tensorcnt)
#define HAVE_TDM 1
#endif
#endif
#ifndef HAVE_TDM
#define HAVE_TDM 0
#endif

// ---------------------------------------------------------------------------
// Kernel 1: nh[N,32] = x[N,3] @ v[3,32] via V_WMMA_F32_16X16X4_F32 (K padded 3->4)
// One wave per 16-node tile; two WMMAs cover t=0..15 and t=16..31. Full f32.
// ---------------------------------------------------------------------------
__global__ void nh_wmma_kernel(const float* __restrict__ x,
                               const float* __restrict__ v,
                               float* __restrict__ nh,
                               int n_tiles) {
  const int wave = blockIdx.x * (blockDim.x >> 5) + (threadIdx.x >> 5);
  if (wave >= n_tiles) return;                 // wave-uniform guard
  const int lane = threadIdx.x & 31;
  const int half = lane >> 4;
  const int l    = lane & 15;
  const int base = wave * 16;

  // A matrix (16x4 f32): lanes 0-15 hold {K0,K1}, lanes 16-31 hold {K2,K3(=0)}
  v2f a;
  const float* xr = x + (size_t)(base + l) * N_FEAT;
  if (half == 0) { a.x = xr[0]; a.y = xr[1]; }
  else           { a.x = xr[2]; a.y = 0.0f;  }

#pragma unroll
  for (int c = 0; c < 2; ++c) {                // two 16-column halves of T
    const int tcol = l + 16 * c;
    v2f b;
    if (half == 0) { b.x = v[0 * N_T + tcol]; b.y = v[1 * N_T + tcol]; }
    else           { b.x = v[2 * N_T + tcol]; b.y = 0.0f;              }
    v8f cz = {0.f, 0.f, 0.f, 0.f, 0.f, 0.f, 0.f, 0.f};
    v8f d = __builtin_amdgcn_wmma_f32_16x16x4_f32(
        false, a, false, b, (short)0, cz, false, false);
#pragma unroll
    for (int r = 0; r < 8; ++r) {
      const int node = base + r + 8 * half;
      nh[(size_t)node * N_T + tcol] = d[r];
    }
  }
}

// ---------------------------------------------------------------------------
// Edge bucketing: histogram -> exclusive scan -> scatter (groups edges by graph)
// ---------------------------------------------------------------------------
__global__ void zero_kernel(int* __restrict__ p, int n) {
  int i = blockIdx.x * blockDim.x + threadIdx.x;
  if (i < n) p[i] = 0;
}

__global__ void edge_hist_kernel(const int* __restrict__ ei,
                                 const int* __restrict__ batch,
                                 int* __restrict__ cnt, int n_edges) {
  int e = blockIdx.x * blockDim.x + threadIdx.x;
  if (e < n_edges) {
    int g = batch[ei[e]];                      // batch[edge_index[0][e]]
    atomicAdd(&cnt[g], 1);
  }
}

__global__ void scan_kernel(const int* __restrict__ cnt,
                            int* __restrict__ offs,
                            int* __restrict__ cursor) {
  if (threadIdx.x == 0 && blockIdx.x == 0) {
    int run = 0;
    for (int g = 0; g < N_GRAPHS; ++g) {
      offs[g]   = run;
      cursor[g] = run;
      run += cnt[g];
    }
    offs[N_GRAPHS] = run;
  }
}

__global__ void edge_scatter_kernel(const int* __restrict__ ei,
                                    const int* __restrict__ batch,
                                    int* __restrict__ cursor,
                                    int* __restrict__ bucket, int n_edges) {
  int e = blockIdx.x * blockDim.x + threadIdx.x;
  if (e < n_edges) {
    int g = batch[ei[e]];
    int pos = atomicAdd(&cursor[g], 1);
    bucket[pos] = e;
  }
}

// ---------------------------------------------------------------------------
// TDM helper: DMA `rows` consecutive 32-float nh rows (contiguous) into LDS.
// Descriptor packed per CDNA5 ISA ch.8 (D# groups 0/1; 2D tensor, groups 2/3
// zero). data_size=4B, tile_dim0=32 elems (128B rows, mult of 4B OK).
// Issued by one wave only (TDM ignores EXEC; done tracked with TENSORcnt).
// ---------------------------------------------------------------------------
#if HAVE_TDM
__device__ __forceinline__ void tdm_load_rows_to_lds(unsigned lds_off,
                                                     const float* gptr,
                                                     int rows) {
  const unsigned long long ga = (unsigned long long)(uintptr_t)gptr;
  u32x4 g0;
  g0.x = 1u;                                           // count=1, user mode
  g0.y = lds_off;                                      // lds_addr (bytes)
  g0.z = (unsigned)ga;                                 // global_addr[31:0]
  g0.w = (unsigned)((ga >> 32) & 0x01FFFFFFu)          // global_addr[56:32]
       | (2u << 30);                                   // type=2 ("image")
  i32x8 g1;
  g1[0] = 0x00020000;                                  // data_size=2 -> 4 bytes
  g1[1] = (int)(N_T << 16);                            // tensor_dim0 = 32
  g1[2] = (int)((rows & 0xFFFF) << 16);                // tensor_dim1 = rows
  g1[3] = (int)(N_T << 16);                            // tile_dim0 = 32
  g1[4] = (rows & 0xFFFF);                             // tile_dim1 = rows
  g1[5] = N_T;                                         // tensor_dim0_stride = 32
  g1[6] = 0;
  g1[7] = 0;
  i32x4 gz = {0, 0, 0, 0};
#if __clang_major__ >= 23
  i32x8 gz8 = {0, 0, 0, 0, 0, 0, 0, 0};
  __builtin_amdgcn_tensor_load_to_lds(g0, g1, gz, gz, gz8, 0);
#else
  __builtin_amdgcn_tensor_load_to_lds(g0, g1, gz, gz, 0);
#endif
}
#endif

// ---------------------------------------------------------------------------
// Main kernel: one block per graph, 1024 threads = 32 waves (wave = s, lane = t).
// Node chunks are DMA'd to LDS via the Tensor Data Mover; edge endpoint rows
// are gathered to LDS manually. sigmoid via v_exp_f32 (base-2) + v_rcp_f32.
// out[s][g][t] = sum_nodes sigmoid - sum_edges sigmoid.
// ---------------------------------------------------------------------------
#define CHUNK 64

__device__ __forceinline__ float fast_sigmoid(float z) {
  // sigmoid(z) = 1 / (1 + 2^(-z*log2(e)))
  float e = __builtin_amdgcn_exp2f(z * -1.4426950408889634f);
  return __builtin_amdgcn_rcpf(1.0f + e);
}

__global__ __launch_bounds__(1024)
void ecc_kernel(const float* __restrict__ nh,
                const int* __restrict__ ei,
                const int* __restrict__ batch,
                const float* __restrict__ lin,
                const int* __restrict__ offs,
                const int* __restrict__ bucket,
                float* __restrict__ out) {
  __shared__ float sbuf[CHUNK * 2 * N_T];   // 16 KB: nodes use half, edges use all
  __shared__ int   snodes[CHUNK * 2];

  const int g    = blockIdx.x;
  const int s    = threadIdx.x >> 5;
  const int lane = threadIdx.x & 31;
  const float thr = lin[s];

  // node range for graph g: batch is sorted, binary search lower bounds
  int lo = 0, hi = N_NODES;
  while (lo < hi) { int m = (lo + hi) >> 1; if (batch[m] < g)     lo = m + 1; else hi = m; }
  const int nstart = lo;
  lo = nstart; hi = N_NODES;
  while (lo < hi) { int m = (lo + hi) >> 1; if (batch[m] < g + 1) lo = m + 1; else hi = m; }
  const int nend = lo;

  // ---- node ECC ----
  float acc_n = 0.0f;
  for (int base = nstart; base < nend; base += CHUNK) {
    const int cnt = min(CHUNK, nend - base);
#if HAVE_TDM
    if (threadIdx.x < 32) {                  // one wave drives the DMA engine
      tdm_load_rows_to_lds((unsigned)(uintptr_t)&sbuf[0],
                           nh + (size_t)base * N_T, cnt);
      __builtin_amdgcn_s_wait_tensorcnt(0);
    }
#else
    for (int idx = threadIdx.x; idx < cnt * N_T; idx += blockDim.x)
      sbuf[idx] = nh[(size_t)base * N_T + idx];
#endif
    __syncthreads();
    for (int i = 0; i < cnt; ++i) {
      float h = sbuf[i * N_T + lane];        // bank-conflict-free: banks (32i+t)%64
      acc_n += fast_sigmoid(200.0f * (thr - h));
    }
    __syncthreads();
  }

  // ---- edge ECC ----
  const int estart = offs[g];
  const int eend   = offs[g + 1];
  float acc_e = 0.0f;
  for (int base = estart; base < eend; base += CHUNK) {
    const int cnt = min(CHUNK, eend - base);
    // phase 1: resolve endpoint node ids for this chunk
    if (threadIdx.x < 2 * cnt) {
      int j = threadIdx.x >> 1;
      int r = threadIdx.x & 1;
      int e = bucket[base + j];
      snodes[threadIdx.x] = ei[r * N_EDGES + e];
    }
    __syncthreads();
    // phase 2: stage both endpoint nh rows (coalesced 128B row reads)
    for (int idx = threadIdx.x; idx < cnt * 2 * N_T; idx += blockDim.x) {
      int j = idx >> 6;            // edge within chunk
      int r = (idx >> 5) & 1;      // endpoint
      int t = idx & 31;
      int node = snodes[(j << 1) | r];
      sbuf[idx] = nh[(size_t)node * N_T + t];
    }
    __syncthreads();
    for (int j = 0; j < cnt; ++j) {
      float h0 = sbuf[j * 2 * N_T + lane];
      float h1 = sbuf[j * 2 * N_T + N_T + lane];
      float h  = fmaxf(h0, h1);
      acc_e += fast_sigmoid(200.0f * (thr - h));
    }
    __syncthreads();
  }

  out[(size_t)s * N_GRAPHS * N_T + (size_t)g * N_T + lane] = acc_n - acc_e;
}

// ---------------------------------------------------------------------------
// Launch
// ---------------------------------------------------------------------------
extern "C" void kernel_launch(void* const* d_in, const int* in_sizes, int n_in,
                              void* d_out, int out_size, void* d_ws, size_t ws_size,
                              hipStream_t stream) {
  const float* x     = (const float*)d_in[0];
  const int*   ei    = (const int*)d_in[1];   // edge_index [2, E]
  const int*   batch = (const int*)d_in[2];   // sorted graph ids [N]
  const float* v     = (const float*)d_in[3]; // [3, 32]
  const float* lin   = (const float*)d_in[4]; // [32]
  float*       out   = (float*)d_out;         // [S, G, T]

  // workspace carve-up (all 256B aligned)
  uint8_t* ws = (uint8_t*)d_ws;
  size_t o = 0;
  float* nh = (float*)(ws + o);  o += (size_t)N_NODES * N_T * sizeof(float);   // 6.4 MB
  o = (o + 255) & ~(size_t)255;
  int* cnt    = (int*)(ws + o);  o += 256;                                     // G ints
  int* cursor = (int*)(ws + o);  o += 256;
  int* offs   = (int*)(ws + o);  o += 1024;                                    // G+1 ints
  int* bucket = (int*)(ws + o);  o += (size_t)N_EDGES * sizeof(int);           // 400 KB

  // 1) nh = x @ v  (WMMA f32 16x16x4, 3125 tiles of 16 nodes, 8 waves/block)
  const int n_tiles = N_NODES / 16;                 // 50000/16 = 3125 exact
  nh_wmma_kernel<<<(n_tiles + 7) / 8, 256, 0, stream>>>(x, v, nh, n_tiles);

  // 2) bucket edges by graph
  zero_kernel<<<1, 256, 0, stream>>>(cnt, N_GRAPHS);
  edge_hist_kernel<<<(N_EDGES + 255) / 256, 256, 0, stream>>>(ei, batch, cnt, N_EDGES);
  scan_kernel<<<1, 32, 0, stream>>>(cnt, offs, cursor);
  edge_scatter_kernel<<<(N_EDGES + 255) / 256, 256, 0, stream>>>(ei, batch, cursor, bucket, N_EDGES);

  // 3) fused node+edge ECC accumulation, one block per graph
  ecc_kernel<<<N_GRAPHS, 1024, 0, stream>>>(nh, ei, batch, lin, offs, bucket, out);
}